// SimpleGNN_68891275427856
// MI455X (gfx1250) — compile-verified
//
#include <hip/hip_runtime.h>
#include <math.h>

#define N_CELL    150000
#define N_NET     150000
#define N_PIN     800000
#define N_NETPAIR 400000
#define N_FATHER  150000

typedef float v2f __attribute__((ext_vector_type(2)));
typedef float v8f __attribute__((ext_vector_type(8)));

// D = A(16x4) * B(4x16) + C, fp32 WMMA (CDNA5 V_WMMA_F32_16X16X4_F32)
__device__ __forceinline__ v8f wmma4(v2f a, v2f b, v8f c) {
  return __builtin_amdgcn_wmma_f32_16x16x4_f32(false, a, false, b, (short)0, c,
                                               false, false);
}

__device__ __forceinline__ float softplus_f(float x) {
  return (x > 20.f) ? x : log1pf(expf(x));
}
__device__ __forceinline__ float ssp_f(float x) {
  return softplus_f(x) - 0.6931471805599453f;
}

// ---------------------------------------------------------------------------
// Generic node-level GEMM: Y[n x DO] = act( (X * rowScale?) @ W + b? )
// One wave -> 16x16 tile; waves tile DO; block processes TPB row-tiles so the
// LDS-resident weights are reused TPB times (amortize L2->LDS staging).
// ACT: 0=linear 1=tanh 2=ssp
// ---------------------------------------------------------------------------
template <int DI, int DO, int ACT, bool HAS_BIAS, bool HAS_SCALE, int TPB>
__global__ __launch_bounds__(DO * 2) void gemm_node_k(
    const float* __restrict__ X, const float* __restrict__ W,
    const float* __restrict__ Bv, const float* __restrict__ rowScale,
    float* __restrict__ Y, int n) {
  __shared__ float sW[DI * DO];
  __shared__ float sB[DO];
  for (int i = threadIdx.x; i < DI * DO; i += blockDim.x) sW[i] = W[i];
  if (HAS_BIAS)
    for (int i = threadIdx.x; i < DO; i += blockDim.x) sB[i] = Bv[i];
  __syncthreads();

  const int lane = threadIdx.x & 31;
  const int wave = threadIdx.x >> 5;
  const int n0 = wave * 16;
  const int l15 = lane & 15;
  const int hi = lane >> 4;
  const int kh = hi * 2;

#pragma unroll 1
  for (int t = 0; t < TPB; ++t) {
    const int m0 = (blockIdx.x * TPB + t) * 16;
    if (m0 >= n) break;

    // prefetch next tile's A rows (global_prefetch_b8) under this tile's math
    {
      int rowP = m0 + 16 + l15;
      if (rowP > n - 1) rowP = n - 1;
      __builtin_prefetch(X + (size_t)rowP * DI, 0, 0);
    }

    int rowA = m0 + l15;
    if (rowA > n - 1) rowA = n - 1;
    const float scl = HAS_SCALE ? rowScale[rowA] : 1.0f;
    const float* xr = X + (size_t)rowA * DI;

    v8f acc = {};
#pragma unroll
    for (int k = 0; k < DI; k += 4) {
      const int kb = k + kh;
      v2f a;
      a.x = xr[kb] * scl;
      a.y = xr[kb + 1] * scl;
      v2f b;
      b.x = sW[kb * DO + n0 + l15];
      b.y = sW[(kb + 1) * DO + n0 + l15];
      acc = wmma4(a, b, acc);
    }

    const int col = n0 + l15;
    const float bias = HAS_BIAS ? sB[col] : 0.f;
    const int rb = m0 + hi * 8;
#pragma unroll
    for (int v = 0; v < 8; ++v) {
      int row = rb + v;
      if (row < n) {
        float val = acc[v] + bias;
        if (ACT == 1) val = tanhf(val);
        else if (ACT == 2) val = ssp_f(val);
        Y[(size_t)row * DO + col] = val;
      }
    }
  }
}

// ---------------------------------------------------------------------------
// Fused cf_conv edge pipeline; 8 tiles of 16 pins per block (4 waves):
//   t  = ssp(hp_tile(16x32) @ We1 + be1)          (WMMA, result -> LDS)
//   he = ssp(t(16x64) @ We2 + be2)                (WMMA, A from LDS)
//   agg[pin_cell] += he * hv[pin_net]             (atomic scatter)
// 'he' (800k x 64) never touches HBM; weights staged to LDS once per block.
// ---------------------------------------------------------------------------
#define CF_TPB 8
__global__ __launch_bounds__(128) void cf_edge_k(
    const float* __restrict__ hp, const float* __restrict__ hv,
    const int* __restrict__ pairs, const float* __restrict__ We1,
    const float* __restrict__ be1, const float* __restrict__ We2,
    const float* __restrict__ be2, float* __restrict__ agg, int nPin) {
  __shared__ float sW1[32 * 64];
  __shared__ float sW2[64 * 64];
  __shared__ float sB1[64];
  __shared__ float sB2[64];
  __shared__ float sT[16 * 65];  // padded stride: no LDS bank conflicts

  for (int i = threadIdx.x; i < 32 * 64; i += 128) sW1[i] = We1[i];
  for (int i = threadIdx.x; i < 64 * 64; i += 128) sW2[i] = We2[i];
  if (threadIdx.x < 64) {
    sB1[threadIdx.x] = be1[threadIdx.x];
    sB2[threadIdx.x] = be2[threadIdx.x];
  }

  const int lane = threadIdx.x & 31;
  const int wave = threadIdx.x >> 5;
  const int n0 = wave * 16;
  const int l15 = lane & 15;
  const int hi = lane >> 4;
  const int kh = hi * 2;

#pragma unroll 1
  for (int t = 0; t < CF_TPB; ++t) {
    const int e0 = (blockIdx.x * CF_TPB + t) * 16;
    if (e0 >= nPin) break;

    // barrier: weights ready (t==0) / previous tile's sT readers done (t>0)
    __syncthreads();

    // prefetch next tile's hp rows
    {
      int pp = e0 + 16 + l15;
      if (pp > nPin - 1) pp = nPin - 1;
      __builtin_prefetch(hp + (size_t)pp * 32, 0, 0);
    }

    {  // GEMM1: 16x32 @ 32x64
      int p = e0 + l15;
      if (p > nPin - 1) p = nPin - 1;
      const float* xr = hp + (size_t)p * 32;
      v8f acc = {};
#pragma unroll
      for (int k = 0; k < 32; k += 4) {
        int kb = k + kh;
        v2f a;
        a.x = xr[kb];
        a.y = xr[kb + 1];
        v2f b;
        b.x = sW1[kb * 64 + n0 + l15];
        b.y = sW1[(kb + 1) * 64 + n0 + l15];
        acc = wmma4(a, b, acc);
      }
      int col = n0 + l15;
      int rb = hi * 8;
#pragma unroll
      for (int v = 0; v < 8; ++v)
        sT[(rb + v) * 65 + col] = ssp_f(acc[v] + sB1[col]);
    }
    __syncthreads();

    {  // GEMM2: 16x64 @ 64x64, fused multiply + scatter
      v8f acc = {};
#pragma unroll
      for (int k = 0; k < 64; k += 4) {
        int kb = k + kh;
        v2f a;
        a.x = sT[l15 * 65 + kb];
        a.y = sT[l15 * 65 + kb + 1];
        v2f b;
        b.x = sW2[kb * 64 + n0 + l15];
        b.y = sW2[(kb + 1) * 64 + n0 + l15];
        acc = wmma4(a, b, acc);
      }
      int col = n0 + l15;
#pragma unroll
      for (int v = 0; v < 8; ++v) {
        int rl = hi * 8 + v;
        int p = e0 + rl;
        if (p < nPin) {
          int net = pairs[2 * p];
          int cell = pairs[2 * p + 1];
          float he = ssp_f(acc[v] + sB2[col]);
          float val = he * hv[(size_t)net * 64 + col];
          atomicAdd(&agg[(size_t)cell * 64 + col], val);
        }
      }
    }
  }
}

// ---------------------------------------------------------------------------
// Graph-conv helpers
// ---------------------------------------------------------------------------
__global__ void count_edges_k(const int* __restrict__ pairs, int nE,
                              float* __restrict__ c0, float* __restrict__ c1) {
  int tid = blockIdx.x * blockDim.x + threadIdx.x;
  if (tid >= nE) return;
  atomicAdd(&c0[pairs[2 * tid]], 1.f);
  atomicAdd(&c1[pairs[2 * tid + 1]], 1.f);
}

__global__ void inv_sqrt_k(float* __restrict__ v, int n) {
  int tid = blockIdx.x * blockDim.x + threadIdx.x;
  if (tid < n) v[tid] = rsqrtf(fmaxf(v[tid], 1.f));
}

// agg[dst] += h[src]  (64 floats per edge; float4 per thread)
__global__ void scatter_rows_k(const float* __restrict__ h,
                               const int* __restrict__ pairs, int srcCol,
                               int dstCol, int nE, float* __restrict__ agg) {
  int tid = blockIdx.x * blockDim.x + threadIdx.x;
  if (tid >= nE * 16) return;
  int e = tid >> 4, q = tid & 15;
  int s = pairs[2 * e + srcCol];
  int d = pairs[2 * e + dstCol];
  const float4 v = *(const float4*)(h + (size_t)s * 64 + q * 4);
  float* dst = agg + (size_t)d * 64 + q * 4;
  atomicAdd(dst + 0, v.x);
  atomicAdd(dst + 1, v.y);
  atomicAdd(dst + 2, v.z);
  atomicAdd(dst + 3, v.w);
}

// R = agg*iscIn + b   (MAXMODE: R = max(R, that))
template <bool MAXMODE>
__global__ void gc_finalize_k(const float* __restrict__ agg,
                              const float* __restrict__ iscIn,
                              const float* __restrict__ bias,
                              float* __restrict__ R, int n) {
  int tid = blockIdx.x * blockDim.x + threadIdx.x;
  if (tid >= n * 64) return;
  int i = tid >> 6, c = tid & 63;
  float val = agg[tid] * iscIn[i] + bias[c];
  if (MAXMODE) R[tid] = fmaxf(R[tid], val);
  else R[tid] = val;
}

// ---------------------------------------------------------------------------
// Heads: one wave per pair / pin, dual dot-product + shuffle reduction
// ---------------------------------------------------------------------------
__global__ __launch_bounds__(256) void net_head_k(
    const float* __restrict__ hn, const int* __restrict__ nn,
    const float* __restrict__ Wd, const float* __restrict__ bd,
    const float* __restrict__ Wa, const float* __restrict__ ba,
    float* __restrict__ oDis, float* __restrict__ oAng, int nPair) {
  int gw = (blockIdx.x * 256 + threadIdx.x) >> 5;
  int lane = threadIdx.x & 31;
  if (gw >= nPair) return;
  int ra = nn[2 * gw], rb = nn[2 * gw + 1];
  const float* base = (lane < 16) ? (hn + (size_t)ra * 64 + lane * 4)
                                  : (hn + (size_t)rb * 64 + (lane - 16) * 4);
  float4 x = *(const float4*)base;
  float4 wd = *(const float4*)(Wd + lane * 4);
  float4 wa = *(const float4*)(Wa + lane * 4);
  float sd = x.x * wd.x + x.y * wd.y + x.z * wd.z + x.w * wd.w;
  float sa = x.x * wa.x + x.y * wa.y + x.z * wa.z + x.w * wa.w;
  for (int off = 16; off > 0; off >>= 1) {
    sd += __shfl_xor(sd, off, 32);
    sa += __shfl_xor(sa, off, 32);
  }
  if (lane == 0) {
    oDis[gw] = softplus_f(sd + bd[0]);
    oAng[gw] = 2.f / (1.f + expf(-(sa + ba[0])));
  }
}

__global__ __launch_bounds__(256) void pin_head_k(
    const float* __restrict__ hn, const float* __restrict__ hp,
    const float* __restrict__ hc, const int* __restrict__ pairs,
    const float* __restrict__ Wd, const float* __restrict__ bd,
    const float* __restrict__ Wa, const float* __restrict__ ba,
    float* __restrict__ oDis, float* __restrict__ oAng, int nPin) {
  int gw = (blockIdx.x * 256 + threadIdx.x) >> 5;
  int lane = threadIdx.x & 31;
  if (gw >= nPin) return;
  int net = pairs[2 * gw], cell = pairs[2 * gw + 1];
  float sd = 0.f, sa = 0.f;
  for (int k = lane; k < 160; k += 32) {
    float x = (k < 64) ? hn[(size_t)net * 64 + k]
              : (k < 96) ? hp[(size_t)gw * 32 + (k - 64)]
                         : hc[(size_t)cell * 64 + (k - 96)];
    sd += x * Wd[k];
    sa += x * Wa[k];
  }
  for (int off = 16; off > 0; off >>= 1) {
    sd += __shfl_xor(sd, off, 32);
    sa += __shfl_xor(sa, off, 32);
  }
  if (lane == 0) {
    oDis[gw] = softplus_f(sd + bd[0]);
    oAng[gw] = 2.f / (1.f + expf(-(sa + ba[0])));
  }
}

// ---------------------------------------------------------------------------
extern "C" void kernel_launch(void* const* d_in, const int* in_sizes, int n_in,
                              void* d_out, int out_size, void* d_ws,
                              size_t ws_size, hipStream_t stream) {
  (void)in_sizes; (void)n_in; (void)out_size; (void)ws_size;
  auto F = [&](int i) { return (const float*)d_in[i]; };
  auto I = [&](int i) { return (const int*)d_in[i]; };

  const float* cell_feat = F(0);
  const float* net_feat = F(1);
  const float* pin_feat = F(2);
  const int* pin_pairs = I(3);
  const int* nn_pairs = I(4);
  const int* father = I(5);
  const float *cell_W = F(6), *cell_b = F(7);
  const float *net_W = F(8), *net_b = F(9);
  const float *pin_W = F(10), *pin_b = F(11);
  // layer l params at index 12 + 14*l (Wp,bp,Wf,bf,Ws,bs,Wn,bn,We1,be1,We2,be2,Wo,bo)
  // heads at 54..61 (ndis_W,ndis_b,nang_W,nang_b,pdis_W,pdis_b,pang_W,pang_b)

  float* p = (float*)d_ws;
  auto take = [&](size_t nelem) { float* r = p; p += nelem; return r; };
  float* hc0 = take((size_t)N_CELL * 64);
  float* hc1 = take((size_t)N_CELL * 64);
  float* hn0 = take((size_t)N_NET * 64);
  float* hn1 = take((size_t)N_NET * 64);
  float* hp = take((size_t)N_PIN * 32);
  float* tmp = take((size_t)N_NET * 64);  // hv / graph_conv GEMM output
  float* agg = take((size_t)N_NET * 64);  // scatter accumulator
  float* cnt = take((size_t)4 * N_NET);   // degree -> inv-sqrt in place
  float* isc_pn = cnt;                    // 1/sqrt(deg over pin_net)
  float* isc_pc = cnt + N_NET;            // 1/sqrt(deg over pin_cell)
  float* isc_fs = cnt + 2 * N_NET;        // 1/sqrt(deg over f_src)
  float* isc_fd = cnt + 3 * N_NET;        // 1/sqrt(deg over f_dst)

  const int NT_NODE = N_CELL / 16;               // 9375 row tiles
  const int NB_NODE = (NT_NODE + 3) / 4;         // TPB=4
  const int NT_PIN = N_PIN / 16;                 // 50000 row tiles
  const int NB_PIN = (NT_PIN + 7) / 8;           // TPB=8
  const int NB_CF = (NT_PIN + CF_TPB - 1) / CF_TPB;

  // --- degrees (constant across layers) ---
  hipMemsetAsync(cnt, 0, (size_t)4 * N_NET * sizeof(float), stream);
  count_edges_k<<<(N_PIN + 255) / 256, 256, 0, stream>>>(pin_pairs, N_PIN,
                                                         isc_pn, isc_pc);
  count_edges_k<<<(N_FATHER + 255) / 256, 256, 0, stream>>>(father, N_FATHER,
                                                            isc_fs, isc_fd);
  inv_sqrt_k<<<(4 * N_NET + 255) / 256, 256, 0, stream>>>(cnt, 4 * N_NET);

  // --- input embeddings (tanh) ---
  gemm_node_k<16, 64, 1, true, false, 4><<<NB_NODE, 128, 0, stream>>>(
      cell_feat, cell_W, cell_b, nullptr, hc0, N_CELL);
  gemm_node_k<16, 64, 1, true, false, 4><<<NB_NODE, 128, 0, stream>>>(
      net_feat, net_W, net_b, nullptr, hn0, N_NET);
  gemm_node_k<8, 32, 1, true, false, 8><<<NB_PIN, 64, 0, stream>>>(
      pin_feat, pin_W, pin_b, nullptr, hp, N_PIN);

  float *hc_in = hc0, *hc_out = hc1;
  float *hn_in = hn0, *hn_out = hn1;

  for (int l = 0; l < 3; ++l) {
    const int base = 12 + 14 * l;
    const float *Wp = F(base + 0), *bp = F(base + 1);
    const float *Wf = F(base + 2), *bf = F(base + 3);
    const float *Ws = F(base + 4), *bs = F(base + 5);
    const float *Wn = F(base + 6), *bn = F(base + 7);
    const float *We1 = F(base + 8), *be1 = F(base + 9);
    const float *We2 = F(base + 10), *be2 = F(base + 11);
    const float *Wo = F(base + 12), *bo = F(base + 13);

    // cf_conv -> new hc
    gemm_node_k<64, 64, 0, true, false, 4><<<NB_NODE, 128, 0, stream>>>(
        hn_in, Wn, bn, nullptr, tmp, N_NET);  // hv = hn@Wn + bn
    hipMemsetAsync(agg, 0, (size_t)N_CELL * 64 * sizeof(float), stream);
    cf_edge_k<<<NB_CF, 128, 0, stream>>>(hp, tmp, pin_pairs, We1, be1, We2,
                                         be2, agg, N_PIN);
    gemm_node_k<64, 64, 2, true, false, 4><<<NB_NODE, 128, 0, stream>>>(
        agg, Wo, bo, nullptr, hc_out, N_CELL);  // hc' = ssp(agg@Wo + bo)

    // r_pins: cells -> nets (writes hn_out)
    gemm_node_k<64, 64, 0, false, true, 4><<<NB_NODE, 128, 0, stream>>>(
        hc_in, Wp, nullptr, isc_pc, tmp, N_CELL);
    hipMemsetAsync(agg, 0, (size_t)N_NET * 64 * sizeof(float), stream);
    scatter_rows_k<<<(N_PIN * 16 + 255) / 256, 256, 0, stream>>>(
        tmp, pin_pairs, 1, 0, N_PIN, agg);
    gc_finalize_k<false><<<(N_NET * 64 + 255) / 256, 256, 0, stream>>>(
        agg, isc_pn, bp, hn_out, N_NET);

    // r_f: f_src -> f_dst (max into hn_out)
    gemm_node_k<64, 64, 0, false, true, 4><<<NB_NODE, 128, 0, stream>>>(
        hn_in, Wf, nullptr, isc_fs, tmp, N_NET);
    hipMemsetAsync(agg, 0, (size_t)N_NET * 64 * sizeof(float), stream);
    scatter_rows_k<<<(N_FATHER * 16 + 255) / 256, 256, 0, stream>>>(
        tmp, father, 0, 1, N_FATHER, agg);
    gc_finalize_k<true><<<(N_NET * 64 + 255) / 256, 256, 0, stream>>>(
        agg, isc_fd, bf, hn_out, N_NET);

    // r_s: f_dst -> f_src (max into hn_out)
    gemm_node_k<64, 64, 0, false, true, 4><<<NB_NODE, 128, 0, stream>>>(
        hn_in, Ws, nullptr, isc_fd, tmp, N_NET);
    hipMemsetAsync(agg, 0, (size_t)N_NET * 64 * sizeof(float), stream);
    scatter_rows_k<<<(N_FATHER * 16 + 255) / 256, 256, 0, stream>>>(
        tmp, father, 1, 0, N_FATHER, agg);
    gc_finalize_k<true><<<(N_NET * 64 + 255) / 256, 256, 0, stream>>>(
        agg, isc_fs, bs, hn_out, N_NET);

    float* t2;
    t2 = hc_in; hc_in = hc_out; hc_out = t2;
    t2 = hn_in; hn_in = hn_out; hn_out = t2;
  }

  // --- heads ---
  float* out = (float*)d_out;
  float* net_dis = out;
  float* net_ang = out + N_NETPAIR;
  float* pin_dis = out + 2 * N_NETPAIR;
  float* pin_ang = out + 2 * N_NETPAIR + N_PIN;

  net_head_k<<<(N_NETPAIR * 32 + 255) / 256, 256, 0, stream>>>(
      hn_in, nn_pairs, F(54), F(55), F(56), F(57), net_dis, net_ang, N_NETPAIR);
  pin_head_k<<<(N_PIN * 32 + 255) / 256, 256, 0, stream>>>(
      hn_in, hp, hc_in, pin_pairs, F(58), F(59), F(60), F(61), pin_dis, pin_ang,
      N_PIN);
}